// RGU_1357209665875
// MI455X (gfx1250) — compile-verified
//
#include <hip/hip_runtime.h>

typedef __attribute__((ext_vector_type(2))) float v2f;
typedef __attribute__((ext_vector_type(4))) float v4f;
typedef __attribute__((ext_vector_type(8))) float v8f;
typedef __attribute__((ext_vector_type(4))) int   v4i;

#define NC 512          // N_CHILDREN
#define NH 64           // HIDDEN
#define XS_STRIDE 516   // 516 % 64 == 4 -> conflict-free strided b64 reads, 16B-aligned rows
#define HS_STRIDE 68    // 68 % 64 == 4

// ---------- gfx1250 feature shims (guarded; fall back cleanly) ----------
#if __has_builtin(__builtin_amdgcn_tanhf)
#define TANH_F32(x) __builtin_amdgcn_tanhf(x)   // hardware V_TANH_F32
#else
#define TANH_F32(x) tanhf(x)
#endif

#if __has_builtin(__builtin_amdgcn_global_load_async_to_lds_b128)
#define HAS_ASYNC_LDS 1
#else
#define HAS_ASYNC_LDS 0
#endif

// address-space-qualified int4 types matching the async-LDS builtin's signature
typedef __attribute__((address_space(1))) v4i v4i_glob;   // global (prints as __device__)
typedef __attribute__((address_space(3))) v4i v4i_lds;    // LDS

// copy 16 bytes global -> LDS (async DMA path on gfx1250, VGPR round-trip otherwise)
__device__ __forceinline__ void copy16_g2l(const float* g, float* l) {
#if HAS_ASYNC_LDS
    __builtin_amdgcn_global_load_async_to_lds_b128(
        (v4i_glob*)g, (v4i_lds*)l, 0, 0);
#else
    *(v4f*)l = *(const v4f*)g;
#endif
}

__device__ __forceinline__ void wait_g2l() {
#if HAS_ASYNC_LDS
#if __has_builtin(__builtin_amdgcn_s_wait_asynccnt)
    __builtin_amdgcn_s_wait_asynccnt(0);
#else
    asm volatile("s_wait_asynccnt 0" ::: "memory");
#endif
#endif
}

// ---- one-time weight repack into workspace ----
// W1P[p*128 + 2n + r] = W1[n][2p+r]   (pair-interleaved transpose, k=2p+r, 512x64)
// W1L[n]              = W1[n][512]    (epsilon column)
// W2P[p*1024 + 2n + r] = W2[n][2p+r]  (pair-interleaved transpose, 64x512)
// => every WMMA B fragment {(k,n),(k+1,n)} is one coalesced b64 load.
__global__ void prep_pack(const float* __restrict__ W1,
                          const float* __restrict__ W2,
                          float* __restrict__ W1P,
                          float* __restrict__ W1L,
                          float* __restrict__ W2P) {
    int tid = blockIdx.x * blockDim.x + threadIdx.x;
    if (tid < 512 * 64) {                      // W1P
        int p = tid >> 7, rem = tid & 127;
        int n = rem >> 1, r = rem & 1;
        W1P[tid] = W1[n * 513 + 2 * p + r];
    }
    int t1 = tid - 512 * 64;
    if (t1 >= 0 && t1 < 64) {                  // W1L
        W1L[t1] = W1[t1 * 513 + 512];
    }
    int t2 = tid - (512 * 64 + 64);
    if (t2 >= 0 && t2 < 32 * 1024) {           // W2P
        int p = t2 >> 10, rem = t2 & 1023;
        int n = rem >> 1, r = rem & 1;
        W2P[t2] = W2[n * 64 + 2 * p + r];
    }
}

// ---- fused kernel: one wave32 handles a 16-row tile end-to-end ----
__global__ __launch_bounds__(32)
void fused_reconcile(const float* __restrict__ x,      // (B,512)
                     const float* __restrict__ xcons,  // (B,1)
                     const float* __restrict__ cap,    // (512)
                     const float* __restrict__ b1,     // (64)
                     const float* __restrict__ b2,     // (512)
                     const float* __restrict__ W1P,    // packed 512x64
                     const float* __restrict__ W1L,    // 64
                     const float* __restrict__ W2P,    // packed 64x512
                     float* __restrict__ out) {        // (B,512)
    extern __shared__ float lds[];
    float* cs = lds;                        // 512 floats: cap
    float* hs = cs + 512;                   // 16*68: h1 tile
    float* ss = hs + 16 * HS_STRIDE;        // 16*516: tanh(z) tile
    float* xs = ss + 16 * XS_STRIDE;        // 16*516: x tile

    const int lane = threadIdx.x;
    const int lm   = lane & 15;             // column-within-tile lane id
    const int hf   = lane >> 4;             // half-wave (row / K-pair selector)
    const int r0   = blockIdx.x << 4;       // first row of this tile

    // ---- stage cap + x tile into LDS via async DMA (no VGPR round-trip) ----
#pragma unroll
    for (int i = 0; i < 4; ++i) {
        const int c = 4 * (lane + 32 * i);
        copy16_g2l(cap + c, cs + c);
    }
#pragma unroll
    for (int m = 0; m < 16; ++m) {
        const float* xr = x + (size_t)(r0 + m) * NC;
#pragma unroll
        for (int i = 0; i < 4; ++i) {
            const int c = 4 * (lane + 32 * i);
            copy16_g2l(xr + c, xs + m * XS_STRIDE + c);
        }
    }
    wait_g2l();
    __syncthreads();

    // ---- cap_sum and per-row dot(x,cap) from LDS ----
    v4f cv[4];
    float capsum = 0.f;
#pragma unroll
    for (int i = 0; i < 4; ++i) {
        cv[i] = ((const v4f*)cs)[lane + 32 * i];
        capsum += cv[i].x + cv[i].y + cv[i].z + cv[i].w;
    }
#pragma unroll
    for (int s = 16; s >= 1; s >>= 1) capsum += __shfl_xor(capsum, s, 32);

    float dot[16];
#pragma unroll
    for (int m = 0; m < 16; ++m) {
        const v4f* xr = (const v4f*)(xs + m * XS_STRIDE);
        float d = 0.f;
#pragma unroll
        for (int i = 0; i < 4; ++i) {
            v4f xv = xr[lane + 32 * i];
            d += xv.x * cv[i].x + xv.y * cv[i].y + xv.z * cv[i].z + xv.w * cv[i].w;
        }
        dot[m] = d;
    }
#pragma unroll
    for (int m = 0; m < 16; ++m) {
#pragma unroll
        for (int s = 16; s >= 1; s >>= 1) dot[m] += __shfl_xor(dot[m], s, 32);
    }
    const float inv_capsum = 1.0f / capsum;
    float eps[16];
#pragma unroll
    for (int m = 0; m < 16; ++m) eps[m] = xcons[r0 + m] - dot[m] * inv_capsum;

    // ---- GEMM1: h1 = relu(x @ W1a^T + eps*w1_last + b1), 16x64 in 4 WMMA tiles ----
    v8f C1[4];
#pragma unroll
    for (int t = 0; t < 4; ++t) {
        const int n = t * 16 + lm;
        const float bn = b1[n];
        const float wl = W1L[n];
#pragma unroll
        for (int v = 0; v < 8; ++v) {
            float e = hf ? eps[v + 8] : eps[v];
            C1[t][v] = bn + e * wl;
        }
    }
    const int koff = 2 * hf;
    const v2f* B1 = (const v2f*)W1P;        // B1[p*64 + n] = {W1[n][2p], W1[n][2p+1]}
#pragma unroll 2
    for (int k0 = 0; k0 < NC; k0 += 4) {
        const int p = (k0 >> 1) + hf;       // this half-wave's K pair
        // A fragment (16x4): lane = M row, VGPR0/1 = K kk,kk+1
        v2f a = *(const v2f*)(xs + lm * XS_STRIDE + k0 + koff);
#pragma unroll
        for (int t = 0; t < 4; ++t) {
            v2f b = B1[p * 64 + t * 16 + lm];   // single coalesced b64 B fragment
            C1[t] = __builtin_amdgcn_wmma_f32_16x16x4_f32(
                        false, a, false, b, (short)0, C1[t], false, false);
        }
    }
    // relu + write h1 tile to LDS (C/D layout: VGPR v -> row v or v+8)
#pragma unroll
    for (int t = 0; t < 4; ++t) {
#pragma unroll
        for (int v = 0; v < 8; ++v) {
            float h = C1[t][v];
            h = h > 0.f ? h : 0.f;
            hs[(v + 8 * hf) * HS_STRIDE + t * 16 + lm] = h;
        }
    }
    __syncthreads();

    // ---- GEMM2: z = h1 @ W2^T + b2; hw tanh; row-sum partials; s -> LDS ----
    float psum[8];
#pragma unroll
    for (int v = 0; v < 8; ++v) psum[v] = 0.f;

    const v2f* B2 = (const v2f*)W2P;        // B2[p*512 + n] = {W2[n][2p], W2[n][2p+1]}
    for (int n0 = 0; n0 < NC; n0 += 32) {   // process N-tiles in pairs for ILP
        const float bb0 = b2[n0 + lm];
        const float bb1 = b2[n0 + 16 + lm];
        v8f Z0, Z1;
#pragma unroll
        for (int v = 0; v < 8; ++v) { Z0[v] = bb0; Z1[v] = bb1; }
#pragma unroll
        for (int k0 = 0; k0 < NH; k0 += 4) {
            const int p = (k0 >> 1) + hf;
            v2f a = *(const v2f*)(hs + lm * HS_STRIDE + k0 + koff);
            v2f b0  = B2[p * 512 + n0 + lm];
            v2f b1v = B2[p * 512 + n0 + 16 + lm];
            Z0 = __builtin_amdgcn_wmma_f32_16x16x4_f32(
                     false, a, false, b0, (short)0, Z0, false, false);
            Z1 = __builtin_amdgcn_wmma_f32_16x16x4_f32(
                     false, a, false, b1v, (short)0, Z1, false, false);
        }
#pragma unroll
        for (int v = 0; v < 8; ++v) {
            float s0 = TANH_F32(Z0[v]);
            float s1 = TANH_F32(Z1[v]);
            psum[v] += s0 + s1;             // both columns belong to the same row
            const int row = v + 8 * hf;
            ss[row * XS_STRIDE + n0 + lm]      = s0;
            ss[row * XS_STRIDE + n0 + 16 + lm] = s1;
        }
    }
    // reduce row sums across the 16 lanes of each half (rows 0-7 | 8-15)
#pragma unroll
    for (int v = 0; v < 8; ++v) {
#pragma unroll
        for (int s = 8; s >= 1; s >>= 1) psum[v] += __shfl_xor(psum[v], s, 32);
    }
    float rowfac[8];
#pragma unroll
    for (int v = 0; v < 8; ++v) {
        float e = hf ? eps[v + 8] : eps[v];
        rowfac[v] = e * capsum / psum[v];   // (eps * cap_sum) / sum(tanh(z))
    }
    __syncthreads();

    // ---- reconcile + coalesced store: out = x + s * rowfac / cap ----
#pragma unroll 2
    for (int n0 = 0; n0 < NC; n0 += 16) {
        const float ic = 1.0f / cs[n0 + lm];
#pragma unroll
        for (int v = 0; v < 8; ++v) {
            const int row = v + 8 * hf;
            float val = xs[row * XS_STRIDE + n0 + lm]
                      + ss[row * XS_STRIDE + n0 + lm] * rowfac[v] * ic;
            out[(size_t)(r0 + row) * NC + n0 + lm] = val;
        }
    }
}

extern "C" void kernel_launch(void* const* d_in, const int* in_sizes, int n_in,
                              void* d_out, int out_size, void* d_ws, size_t ws_size,
                              hipStream_t stream) {
    const float* x_cpm  = (const float*)d_in[0];
    const float* x_cons = (const float*)d_in[1];
    const float* cap    = (const float*)d_in[2];
    const float* W1     = (const float*)d_in[3];
    const float* b1     = (const float*)d_in[4];
    const float* W2     = (const float*)d_in[5];
    const float* b2     = (const float*)d_in[6];
    float* out = (float*)d_out;

    float* W1P = (float*)d_ws;            // 512*64 floats
    float* W1L = W1P + 512 * 64;          // 64 floats
    float* W2P = W1L + 64;                // 32*1024 floats (total ~262 KB of ws)

    const int prep_elems = 512 * 64 + 64 + 32 * 1024;
    prep_pack<<<(prep_elems + 255) / 256, 256, 0, stream>>>(W1, W2, W1P, W1L, W2P);

    const size_t lds_bytes =
        (512 + 16 * HS_STRIDE + 16 * XS_STRIDE + 16 * XS_STRIDE) * sizeof(float);
    fused_reconcile<<<65536 / 16, 32, lds_bytes, stream>>>(
        x_cpm, x_cons, cap, b1, b2, W1P, W1L, W2P, out);
}